// SkeletalConvolution_80307298501385
// MI455X (gfx1250) — compile-verified
//
#include <hip/hip_runtime.h>

typedef __attribute__((ext_vector_type(2))) float v2f;
typedef __attribute__((ext_vector_type(4))) float f4;
typedef __attribute__((ext_vector_type(8))) float v8f;

#define NJ 22   // joints (rows/cols of adjacency)
#define TT 50   // time dimension
#define MP 32   // padded M (2 x 16 tiles)
#define KP 24   // padded K (6 x 4 steps)
#define NP 64   // padded N (4 x 16 tiles)
#define NEDGE 43

// 22 self-links + 21 directed bone links (0-indexed)
__device__ __constant__ unsigned char kRows[NEDGE] = {
    0,1,2,3,4,5,6,7,8,9,10,11,12,13,14,15,16,17,18,19,20,21,
    0,1,2,4,5,6,0,4,8,9,10,9,12,13,14,14,9,17,18,19,19};
__device__ __constant__ unsigned char kCols[NEDGE] = {
    0,1,2,3,4,5,6,7,8,9,10,11,12,13,14,15,16,17,18,19,20,21,
    1,2,3,5,6,7,8,8,9,10,11,12,13,14,15,16,17,18,19,20,21};

// ---------------------------------------------------------------------------
// Kernel 1: bulk zero-fill of the output (169 MB -> ~7.3 us @ 23.3 TB/s).
// Nontemporal 128-bit stores; out_size is a multiple of 4 but guard anyway.
// ---------------------------------------------------------------------------
__global__ void __launch_bounds__(256) skl_zero_kernel(float* __restrict__ out,
                                                       long n_elems) {
    long n4 = n_elems >> 2;
    long i = (long)blockIdx.x * blockDim.x + threadIdx.x;
    if (i < n4) {
        f4 z = {0.f, 0.f, 0.f, 0.f};
        __builtin_nontemporal_store(z, ((f4*)out) + i);
    }
    if (i == 0) {   // scalar tail (none for 42,240,000, but be safe)
        for (long t = n4 << 2; t < n_elems; ++t)
            __builtin_nontemporal_store(0.f, out + t);
    }
}

// ---------------------------------------------------------------------------
// Kernel 2: one workgroup (8 waves) computes the 22x50 nonzero block as
//   OUT(22x50) = S(22x22, 0/1 adjacency) @ X(22x50)
// via V_WMMA_F32_16X16X4_F32 (exact fp32). Each wave owns one 16x16 D tile
// (mi in {0,1}, ni in {0..3}) and accumulates over 6 K-steps of 4.
//
// Fragment layouts (ISA 7.12.2, reg v / lane-half h -> index v + nregs*h):
//   A 16x4 : lane&15 = M, a.x = S[M][k0+2h],   a.y = S[M][k0+2h+1]
//   B 4x16 : lane&15 = N, b.x = X[k0+2h][N],   b.y = X[k0+2h+1][N]
//   D 16x16: reg v, M = v + 8h, N = lane&15
// ---------------------------------------------------------------------------
__global__ void __launch_bounds__(256) skl_wmma_kernel(const float* __restrict__ x,
                                                       float* __restrict__ out) {
    __shared__ float S[MP][KP];   // padded adjacency, row-major
    __shared__ float X[KP][NP];   // padded x block, row-major

    const int tid = threadIdx.x;

    // zero-pad LDS
    for (int i = tid; i < MP * KP; i += 256) (&S[0][0])[i] = 0.f;
    for (int i = tid; i < KP * NP; i += 256) (&X[0][0])[i] = 0.f;
    __syncthreads();

    // scatter the 43 adjacency ones (all (r,c) pairs unique -> plain stores)
    if (tid < NEDGE) S[kRows[tid]][kCols[tid]] = 1.0f;
    // stage the live 22x50 block of x_flat (rows 0..21 of batch 0)
    for (int i = tid; i < NJ * TT; i += 256) {
        int r = i / TT, c = i % TT;
        X[r][c] = x[r * TT + c];
    }
    __syncthreads();

    const int wave = tid >> 5;
    const int lane = tid & 31;
    const int mi = wave >> 2;        // 0..1  (M tile)
    const int ni = wave & 3;         // 0..3  (N tile)
    const int h  = lane >> 4;        // lane half
    const int l  = lane & 15;

    const int m = mi * 16 + l;       // A row for this lane
    const int n = ni * 16 + l;       // B/D column for this lane

    v8f c = {};
#pragma unroll
    for (int k0 = 0; k0 < KP; k0 += 4) {
        v2f a, b;
        a.x = S[m][k0 + 2 * h];
        a.y = S[m][k0 + 2 * h + 1];
        b.x = X[k0 + 2 * h][n];
        b.y = X[k0 + 2 * h + 1][n];
        // (neg_a, A, neg_b, B, c_mod, C, reuse_a, reuse_b)
        c = __builtin_amdgcn_wmma_f32_16x16x4_f32(
                false, a, false, b, (short)0, c, false, false);
    }

    // write back the valid 22x50 region (overwrites zeros from kernel 1)
#pragma unroll
    for (int v = 0; v < 8; ++v) {
        int row = mi * 16 + v + 8 * h;
        if (row < NJ && n < TT) out[row * TT + n] = c[v];
    }
}

// ---------------------------------------------------------------------------
extern "C" void kernel_launch(void* const* d_in, const int* in_sizes, int n_in,
                              void* d_out, int out_size, void* d_ws, size_t ws_size,
                              hipStream_t stream) {
    (void)in_sizes; (void)n_in; (void)d_ws; (void)ws_size;
    const float* x = (const float*)d_in[0];
    float* out = (float*)d_out;

    long n_elems = (long)out_size;              // 42,240,000 fp32
    long n4 = n_elems >> 2;
    int threads = 256;
    unsigned int blocks = (unsigned int)((n4 + threads - 1) / threads);

    skl_zero_kernel<<<blocks, threads, 0, stream>>>(out, n_elems);
    skl_wmma_kernel<<<1, 256, 0, stream>>>(x, out);
}